// VocabParallelEmbeddingWithLoRA_7791070675056
// MI455X (gfx1250) — compile-verified
//
#include <hip/hip_runtime.h>

// ---------------- problem constants (match reference) ----------------
#define VOCAB      32000
#define EMBED      2048
#define RANK       16
#define N_ADAPTERS 8
#define S_TOKENS   16384

// each adapter group padded to a multiple of 16 tokens
#define MAX_TILES  ((S_TOKENS + N_ADAPTERS * 15 + 15) / 16)   // 1032
#define PERM_CAP   (MAX_TILES * 16)                           // 16512

// workspace layout in ints:
//   [0..7]   counts
//   [8..16]  padded group offsets (9 entries)
//   [17..24] scatter cursors
//   [32..32+PERM_CAP) permutation (token index per padded slot, -1 = pad)
#define WS_COUNTS  0
#define WS_OFFS    8
#define WS_CURS    17
#define WS_PERM    32

// columns per wave / per block
#define NT_PER_WAVE   8                       // 8 x 16 = 128 cols per wave
#define COLS_PER_WAVE (NT_PER_WAVE * 16)
#define WAVES_PER_BLK 4
#define COLS_PER_BLK  (COLS_PER_WAVE * WAVES_PER_BLK)   // 512

typedef float v2f __attribute__((ext_vector_type(2)));
typedef float v8f __attribute__((ext_vector_type(8)));

// ---------------- grouping kernels (tiny) ----------------
__global__ void init_ws_kernel(int* __restrict__ wsi) {
    int i = blockIdx.x * blockDim.x + threadIdx.x;
    if (i < PERM_CAP) wsi[WS_PERM + i] = -1;
    if (i < N_ADAPTERS) { wsi[WS_COUNTS + i] = 0; wsi[WS_CURS + i] = 0; }
}

__global__ void count_kernel(const int* __restrict__ adapter_ids, int* __restrict__ wsi) {
    int s = blockIdx.x * blockDim.x + threadIdx.x;
    if (s < S_TOKENS) atomicAdd(&wsi[WS_COUNTS + (adapter_ids[s] & (N_ADAPTERS - 1))], 1);
}

__global__ void offsets_kernel(int* __restrict__ wsi) {
    int off = 0;
    wsi[WS_OFFS] = 0;
    for (int g = 0; g < N_ADAPTERS; ++g) {
        off += ((wsi[WS_COUNTS + g] + 15) >> 4) << 4;  // pad to multiple of 16
        wsi[WS_OFFS + g + 1] = off;
    }
}

__global__ void scatter_kernel(const int* __restrict__ adapter_ids, int* __restrict__ wsi) {
    int s = blockIdx.x * blockDim.x + threadIdx.x;
    if (s < S_TOKENS) {
        int g    = adapter_ids[s] & (N_ADAPTERS - 1);
        int slot = atomicAdd(&wsi[WS_CURS + g], 1);
        wsi[WS_PERM + wsi[WS_OFFS + g] + slot] = s;
    }
}

// ---------------- main fused gather + LoRA WMMA kernel ----------------
// block = 128 threads = 4 waves; grid = (EMBED/512, MAX_TILES)
// wave w handles token-tile blockIdx.y, columns [(blockIdx.x*4+w)*128, +128)
//
// Pad-slot handling: a pad slot inherits row 0's token + vocab id. Its A row
// and C row are then bitwise identical to row 0's, so its D row is bitwise
// identical too, and its store is an idempotent duplicate write of row 0's
// value to row 0's address. No masking anywhere -> single uniform store path.
__global__ __launch_bounds__(128) void lora_embed_wmma_kernel(
    const int*   __restrict__ input_ids,
    const float* __restrict__ weight,   // [VOCAB, EMBED]
    const float* __restrict__ embA,     // [N_ADAPTERS, VOCAB, RANK]
    const float* __restrict__ embB,     // [N_ADAPTERS, EMBED, RANK]
    const int*   __restrict__ wsi,
    float*       __restrict__ out)      // [S, EMBED]
{
    const int tile      = blockIdx.y;
    const int base_slot = tile * 16;
    const int* perm     = wsi + WS_PERM;

    // pads are trailing within a group; tile fully unused iff first slot is pad
    if (perm[base_slot] < 0) return;

    __shared__ int sh_tok[16];   // token index s (pad slots aliased to row 0)
    __shared__ int sh_vid[16];   // masked vocab id

    const int t = threadIdx.x;
    if (t < 16) {
        int s = perm[base_slot + t];
        if (s < 0) s = perm[base_slot];            // pad -> duplicate row 0
        int vid = input_ids[s];
        if (vid > VOCAB - 1 || vid < 0) vid = 0;   // reference's OOV masking
        sh_tok[t] = s;
        sh_vid[t] = vid;
    }
    __syncthreads();

    // adapter group owning this tile (uniform)
    int g = 0;
    #pragma unroll
    for (int gg = 0; gg < N_ADAPTERS; ++gg)
        if (base_slot >= wsi[WS_OFFS + gg] && base_slot < wsi[WS_OFFS + gg + 1]) g = gg;

    const int lane  = t & 31;
    const int wave  = t >> 5;
    const int m     = lane & 15;   // A row / B-C-D column within tile
    const int khalf = lane >> 4;   // lane-half: K pair (A/B) or M half (C/D)

    const int n_base  = (blockIdx.x * WAVES_PER_BLK + wave) * COLS_PER_WAVE;
    const int nbase_m = n_base + m;          // this lane's first output column

    // ---- per-row 32-bit element offsets (SADDR + scaled voffset form) ----
    int woffv[8];                  // weight[woffv[i] + nt*16]  = C gather
    int ooffv[8];                  // out[ooffv[i] + nt*16]     = D store
    #pragma unroll
    for (int i = 0; i < 8; ++i) {
        const int row = i + 8 * khalf;
        woffv[i] = sh_vid[row] * EMBED + nbase_m;
        ooffv[i] = sh_tok[row] * EMBED + nbase_m;
    }

    // ---- A fragments: ISA 32-bit 16x4 A layout ----
    // lane(m,khalf), K-step kk holds K = 4*kk + 2*khalf + {0,1}
    const int aoff = (g * VOCAB + sh_vid[m]) * RANK + 2 * khalf;   // < 4.1M elems
    v2f afrag[4];
    #pragma unroll
    for (int kk = 0; kk < 4; ++kk)
        afrag[kk] = *(const v2f*)(embA + aoff + 4 * kk);

    // B fragment base offset for this lane's column-phase (column = nbase_m + nt*16)
    const int boff = (g * EMBED + nbase_m) * RANK + 2 * khalf;     // < 295K elems

    #pragma unroll
    for (int nt = 0; nt < NT_PER_WAVE; ++nt) {
        const int ncol = nt * 16;            // folds into instruction immediates

        // ---- B fragments: B[k][n] = embB[g, n, k] (same K-pair layout as A) ----
        v2f bfrag[4];
        #pragma unroll
        for (int kk = 0; kk < 4; ++kk)
            bfrag[kk] = *(const v2f*)(embB + boff + ncol * RANK + 4 * kk);

        // ---- C = base embedding gather, in 16x16 f32 C/D layout ----
        v8f acc;
        #pragma unroll
        for (int i = 0; i < 8; ++i)
            acc[i] = weight[woffv[i] + ncol];

        // ---- D = A x B + C, chained over K=16 via 4x V_WMMA_F32_16X16X4_F32 ----
        #pragma unroll
        for (int kk = 0; kk < 4; ++kk)
            acc = __builtin_amdgcn_wmma_f32_16x16x4_f32(
                /*neg_a=*/false, afrag[kk],
                /*neg_b=*/false, bfrag[kk],
                /*c_mod=*/(short)0, acc,
                /*reuse_a=*/false, /*reuse_b=*/false);

        // ---- store: streamed output, non-temporal to preserve L2 for weight ----
        #pragma unroll
        for (int i = 0; i < 8; ++i)
            __builtin_nontemporal_store(acc[i], out + ooffv[i] + ncol);
    }
}

// ---------------- launcher ----------------
extern "C" void kernel_launch(void* const* d_in, const int* in_sizes, int n_in,
                              void* d_out, int out_size, void* d_ws, size_t ws_size,
                              hipStream_t stream) {
    (void)in_sizes; (void)n_in; (void)out_size; (void)ws_size;
    const int*   input_ids   = (const int*)d_in[0];
    const int*   adapter_ids = (const int*)d_in[1];
    const float* weight      = (const float*)d_in[2];
    const float* embA        = (const float*)d_in[3];
    const float* embB        = (const float*)d_in[4];
    float* out = (float*)d_out;
    int*   wsi = (int*)d_ws;

    init_ws_kernel<<<(PERM_CAP + 255) / 256, 256, 0, stream>>>(wsi);
    count_kernel<<<(S_TOKENS + 255) / 256, 256, 0, stream>>>(adapter_ids, wsi);
    offsets_kernel<<<1, 1, 0, stream>>>(wsi);
    scatter_kernel<<<(S_TOKENS + 255) / 256, 256, 0, stream>>>(adapter_ids, wsi);

    dim3 grid(EMBED / COLS_PER_BLK, MAX_TILES);   // (4, 1032)
    lora_embed_wmma_kernel<<<grid, 128, 0, stream>>>(input_ids, weight, embA, embB, wsi, out);
}